// Shared_Unit_NL_36386962932119
// MI455X (gfx1250) — compile-verified
//
#include <hip/hip_runtime.h>

typedef __attribute__((ext_vector_type(2))) float v2f;
typedef __attribute__((ext_vector_type(4))) float v4f;
typedef __attribute__((ext_vector_type(8))) float v8f;

constexpr int kD = 256;                 // feature dim
constexpr int kChunks = kD / 4;         // 64 WMMA K-chunks
constexpr int kRowsPerWave = 16;        // one WMMA tile of rows per wave
constexpr int kWavesPerBlock = 8;       // 256 threads
constexpr int kRowsPerBlock = kRowsPerWave * kWavesPerBlock;  // 128

__global__ __launch_bounds__(256) void shared_unit_wmma_kernel(
    const float* __restrict__ drug_cnn,
    const float* __restrict__ drug_kg,
    const float* __restrict__ w_aa,
    const float* __restrict__ w_ab,
    const float* __restrict__ w_ba,
    const float* __restrict__ w_bb,
    const float* __restrict__ bias_cnn,
    const float* __restrict__ bias_kg,
    float* __restrict__ out_cnn,
    float* __restrict__ out_kg)
{
    // Precomputed per-lane B-fragments for every K-chunk (invariant over rows):
    // lane l -> N = l&15 (output column), K-pair = 2*(l>>4) within the chunk.
    // Column 0 = w_aa+w_ab, column 1 = w_ba+w_bb, columns 2..15 = 0.
    __shared__ __align__(16) v2f   s_btab[kChunks][32];        // 16 KiB
    __shared__ __align__(16) float s_bc[kD];                   // cnn bias
    __shared__ __align__(16) float s_bk[kD];                   // kg bias
    __shared__ __align__(16) float s_scl[kRowsPerBlock * 2];   // {s_cnn, s_kg}/row

    const int tid = threadIdx.x;

    // Stage biases (256 threads == kD).
    s_bc[tid] = bias_cnn[tid];
    s_bk[tid] = bias_kg[tid];

    // Build the B-fragment table: 2048 entries / 256 threads = 8 each.
    #pragma unroll
    for (int j = 0; j < 8; ++j) {
        const int f    = tid + 256 * j;
        const int kk   = f >> 5;
        const int l    = f & 31;
        const int n    = l & 15;
        const int half = l >> 4;
        const int k0   = kk * 4 + half * 2;
        v2f b = {0.0f, 0.0f};
        if (n == 0) {
            b.x = w_aa[k0]     + w_ab[k0];
            b.y = w_aa[k0 + 1] + w_ab[k0 + 1];
        } else if (n == 1) {
            b.x = w_ba[k0]     + w_bb[k0];
            b.y = w_ba[k0 + 1] + w_bb[k0 + 1];
        }
        s_btab[kk][l] = b;
    }
    __syncthreads();

    const int lane = tid & 31;
    const int wave = tid >> 5;
    const int n    = lane & 15;   // row id in A-frag / column id in B,D-frag
    const int half = lane >> 4;   // selects K-pair {0,1} vs {2,3} within a chunk
    const int rowBlock = blockIdx.x * kRowsPerBlock;
    const int rowWave  = rowBlock + wave * kRowsPerWave;

    // ---- Phase 1: s = kg_tile(16xK) x W(Kx16) via chained f32 WMMA ----
    // A layout (16x4 f32): lane (half,n) holds A[M=n, K=2*half + {0,1}].
    const v2f* kgRow2 = (const v2f*)(drug_kg + (size_t)(rowWave + n) * kD);

    v8f acc = {};
    #pragma unroll 8
    for (int kk = 0; kk < kChunks; ++kk) {
        v2f a = __builtin_nontemporal_load(&kgRow2[kk * 2 + half]); // b64, streaming
        v2f b = s_btab[kk][lane];                                   // ds_load_b64
        acc = __builtin_amdgcn_wmma_f32_16x16x4_f32(
            /*neg_a=*/false, a, /*neg_b=*/false, b,
            /*c_mod=*/(short)0, acc, /*reuse_a=*/false, /*reuse_b=*/false);
    }

    // D layout: VGPR v holds D[M=v, N=lane] (lanes 0-15) / D[M=v+8, N=lane-16].
    // Column N=0 (lanes 0,16) = s_cnn; column N=1 (lanes 1,17) = s_kg.
    if (n < 2) {
        const int mbase = wave * kRowsPerWave + half * 8;
        #pragma unroll
        for (int v = 0; v < 8; ++v)
            s_scl[(mbase + v) * 2 + n] = acc[v];
    }
    __syncthreads();

    // ---- Phase 2: out = cnn * s + bias, float4 coalesced streaming ----
    const v4f* cnn4 = (const v4f*)drug_cnn;
    v4f* oc4 = (v4f*)out_cnn;
    v4f* ok4 = (v4f*)out_kg;
    const v4f* bc4 = (const v4f*)s_bc;
    const v4f* bk4 = (const v4f*)s_bk;

    #pragma unroll 4
    for (int r = 0; r < kRowsPerWave; ++r) {
        const int lrow = wave * kRowsPerWave + r;
        const size_t grow = (size_t)(rowBlock + lrow) * (kD / 4);
        const float sc = s_scl[lrow * 2 + 0];
        const float sk = s_scl[lrow * 2 + 1];
        #pragma unroll
        for (int hcol = 0; hcol < 2; ++hcol) {
            const int i = hcol * 32 + lane;          // float4 col index 0..63
            v4f c  = __builtin_nontemporal_load(&cnn4[grow + i]);
            v4f bc = bc4[i];
            v4f bk = bk4[i];
            v4f o1, o2;
            o1.x = __builtin_fmaf(c.x, sc, bc.x);  o2.x = __builtin_fmaf(c.x, sk, bk.x);
            o1.y = __builtin_fmaf(c.y, sc, bc.y);  o2.y = __builtin_fmaf(c.y, sk, bk.y);
            o1.z = __builtin_fmaf(c.z, sc, bc.z);  o2.z = __builtin_fmaf(c.z, sk, bk.z);
            o1.w = __builtin_fmaf(c.w, sc, bc.w);  o2.w = __builtin_fmaf(c.w, sk, bk.w);
            __builtin_nontemporal_store(o1, &oc4[grow + i]);
            __builtin_nontemporal_store(o2, &ok4[grow + i]);
        }
    }
}

extern "C" void kernel_launch(void* const* d_in, const int* in_sizes, int n_in,
                              void* d_out, int out_size, void* d_ws, size_t ws_size,
                              hipStream_t stream) {
    const float* drug_cnn = (const float*)d_in[0];
    const float* drug_kg  = (const float*)d_in[1];
    const float* w_aa     = (const float*)d_in[2];
    const float* w_ab     = (const float*)d_in[3];
    const float* w_ba     = (const float*)d_in[4];
    const float* w_bb     = (const float*)d_in[5];
    const float* bcnn     = (const float*)d_in[6];
    const float* bkg      = (const float*)d_in[7];

    const int B = in_sizes[0] / kD;             // 65536
    float* out_cnn = (float*)d_out;
    float* out_kg  = out_cnn + (size_t)B * kD;  // tuple outputs concatenated

    dim3 grid(B / kRowsPerBlock);               // 512 blocks
    dim3 block(kWavesPerBlock * 32);            // 256 threads = 8 waves
    shared_unit_wmma_kernel<<<grid, block, 0, stream>>>(
        drug_cnn, drug_kg, w_aa, w_ab, w_ba, w_bb, bcnn, bkg, out_cnn, out_kg);
}